// RegionProposalNetwork_17643725652127
// MI455X (gfx1250) — compile-verified
//
#include <hip/hip_runtime.h>
#include <stdint.h>

// ---------------- problem constants ----------------
#define T_ANCH   392832
#define NIMG     8
#define K_PRE    1000
#define K_ALL    5000           // 5 levels * 1000
#define K_PAD    5008           // padded to multiple of 16
#define NTILE    (K_PAD / 16)   // 313 tiles per axis
#define SORT_N   8192
#define MASK_W16 320            // u16 words per mask row (padded)
#define NMS_TH   0.7f
#define BBOX_CLIP 4.135166556742356f
#define IMG_WF   1536.0f
#define IMG_HF   1024.0f
#define MIN_SZ   1e-3f

__constant__ int LVL_OFF[5] = {0, 294912, 368640, 387072, 391680};
__constant__ int LVL_CNT[5] = {294912, 73728, 18432, 4608, 1152};
__constant__ int LVL_W[5]   = {384, 192, 96, 48, 24};
__constant__ int LVL_H[5]   = {256, 128, 64, 32, 16};

typedef float v2f __attribute__((ext_vector_type(2)));
typedef float v8f __attribute__((ext_vector_type(8)));
typedef unsigned int u32x4 __attribute__((ext_vector_type(4)));
typedef int i32x4 __attribute__((ext_vector_type(4)));
typedef int i32x8 __attribute__((ext_vector_type(8)));

struct LevelPtrs { const float* obj[5]; const float* del[5]; };

__device__ __forceinline__ unsigned okey(float f) {
  unsigned x = __float_as_uint(f);
  return (x & 0x80000000u) ? ~x : (x | 0x80000000u);   // larger float -> larger key
}

__device__ __forceinline__ unsigned ballot32(bool b) {
#if __has_builtin(__builtin_amdgcn_ballot_w32)
  return __builtin_amdgcn_ballot_w32(b);
#else
  return (unsigned)__ballot(b);
#endif
}

__device__ __forceinline__ int find_lvl(int t) {
  if (t < 294912) return 0;
  if (t < 368640) return 1;
  if (t < 387072) return 2;
  if (t < 391680) return 3;
  return 4;
}

// ---------------- TDM: 1D tile (nElem dwords) global -> LDS ----------------
#if __has_builtin(__builtin_amdgcn_tensor_load_to_lds)
#define HAVE_TDM 1
__device__ __forceinline__ void tdm_load_dwords(unsigned lds_off,
                                                unsigned long long ga,
                                                unsigned nElem) {
  u32x4 g0;
  g0[0] = 1u;                                       // count=1, user descriptor
  g0[1] = lds_off;                                  // lds_addr (bytes)
  g0[2] = (unsigned)(ga & 0xFFFFFFFFull);           // global_addr[31:0]
  g0[3] = ((unsigned)(ga >> 32) & 0x01FFFFFFu) | 0x80000000u; // addr[56:32] | type=2
  i32x8 g1;
  g1[0] = (int)(2u << 16);                          // data_size=4B, no multicast
  g1[1] = (int)((nElem & 0xFFFFu) << 16);           // tensor_dim0[15:0]
  g1[2] = (int)(((nElem >> 16) & 0xFFFFu) | (1u << 16)); // dim0 hi | tensor_dim1=1
  g1[3] = (int)((nElem & 0xFFFFu) << 16);           // tile_dim0 (fits 16 bits)
  g1[4] = 1;                                        // tile_dim1=1, tile_dim2=0
  g1[5] = (int)nElem;                               // tensor_dim0_stride lo
  g1[6] = 0; g1[7] = 0;
  i32x4 g2 = {0,0,0,0}, g3 = {0,0,0,0};
#if defined(__clang_major__) && (__clang_major__ >= 23)
  i32x8 g4 = {0,0,0,0,0,0,0,0};
  __builtin_amdgcn_tensor_load_to_lds(g0, g1, g2, g3, g4, 0);
#else
  __builtin_amdgcn_tensor_load_to_lds(g0, g1, g2, g3, 0);
#endif
}
#endif

// ---------------- K1: orderable score keys for all (img, anchor) ----------------
__global__ void k_scores(LevelPtrs p, unsigned* __restrict__ keys) {
  int gid = blockIdx.x * blockDim.x + threadIdx.x;
  if (gid >= NIMG * T_ANCH) return;
  int img = gid / T_ANCH, t = gid - img * T_ANCH;
  int l = find_lvl(t);
  int q = t - LVL_OFF[l];
  int a = q % 3, xy = q / 3;
  int w = LVL_W[l], h = LVL_H[l];
  int x = xy % w, y = xy / w;
  float o = p.obj[l][((size_t)(img * 3 + a) * h + y) * w + x];
  keys[gid] = okey(o);
}

// ---------------- K2: exact top-1000 per (img, level) via 3-pass radix ----------------
__global__ void __launch_bounds__(256)
k_select(const unsigned* __restrict__ keys, unsigned* __restrict__ sel_key,
         int* __restrict__ sel_idx) {
  int img = blockIdx.x, l = blockIdx.y, tid = threadIdx.x;
  int off = LVL_OFF[l], n = LVL_CNT[l];
  const unsigned* kk = keys + (size_t)img * T_ANCH + off;

  __shared__ unsigned hist[2048];
  __shared__ unsigned sh_prefix;
  __shared__ int sh_need;
  __shared__ unsigned cGT, cEQ;
  if (tid == 0) { sh_prefix = 0u; sh_need = K_PRE; }
  __syncthreads();

  const int shifts[3] = {21, 10, 0};
  const int nbits[3]  = {11, 11, 10};
  for (int p = 0; p < 3; p++) {
    int shift = shifts[p], nb = nbits[p], bins = 1 << nb;
    for (int b = tid; b < bins; b += 256) hist[b] = 0u;
    __syncthreads();
    unsigned pref = sh_prefix;
    for (int idx = tid; idx < n; idx += 256) {
      unsigned u = kk[idx];
      bool cond = (p == 0) || ((u >> (shift + nb)) == pref);
      if (cond) atomicAdd(&hist[(u >> shift) & (unsigned)(bins - 1)], 1u);
    }
    __syncthreads();
    if (tid == 0) {
      int need = sh_need; unsigned cum = 0; int chosen = 0;
      for (int b = bins - 1; b >= 0; b--) {
        unsigned c = hist[b];
        if (cum + c >= (unsigned)need) { chosen = b; sh_need = need - (int)cum; break; }
        cum += c;
      }
      sh_prefix = (pref << nb) | (unsigned)chosen;
    }
    __syncthreads();
  }
  unsigned thr = sh_prefix;
  int rEq = sh_need;                // how many == thr to take
  if (tid == 0) { cGT = 0u; cEQ = 0u; }
  __syncthreads();

  int obase = img * K_ALL + l * K_PRE;
  for (int idx = tid; idx < n; idx += 256) {
    unsigned u = kk[idx];
    if (u > thr) {
      unsigned pos = atomicAdd(&cGT, 1u);
      if (pos < (unsigned)K_PRE) { sel_key[obase + pos] = u; sel_idx[obase + pos] = off + idx; }
    } else if (u == thr) {
      unsigned e = atomicAdd(&cEQ, 1u);
      if (e < (unsigned)rEq) {
        unsigned pos = (unsigned)(K_PRE - rEq) + e;
        sel_key[obase + pos] = u; sel_idx[obase + pos] = off + idx;
      }
    }
  }
}

// ---------------- K3: decode + clip + validity for the 5000 selected ----------------
__global__ void k_gather(LevelPtrs p, const float* __restrict__ anchors,
                         const unsigned* __restrict__ sel_key,
                         const int* __restrict__ sel_idx,
                         float4* __restrict__ cboxes, unsigned* __restrict__ ckey,
                         int* __restrict__ clvl) {
  int gid = blockIdx.x * blockDim.x + threadIdx.x;
  if (gid >= NIMG * K_ALL) return;
  int img = gid / K_ALL, k = gid - img * K_ALL;
  int l = k / K_PRE;
  int t = sel_idx[gid];
  int q = t - LVL_OFF[l];
  int a = q % 3, xy = q / 3;
  int w = LVL_W[l], h = LVL_H[l];
  int x = xy % w, y = xy / w;
  const float* dl = p.del[l];
  size_t hw = (size_t)h * w;
  size_t base = ((size_t)img * 12 + (size_t)a * 4) * hw + (size_t)y * w + x;
  float dx = dl[base], dy = dl[base + hw];
  float dw = fminf(dl[base + 2 * hw], BBOX_CLIP);
  float dh = fminf(dl[base + 3 * hw], BBOX_CLIP);
  float a0 = anchors[4 * t + 0], a1 = anchors[4 * t + 1];
  float a2 = anchors[4 * t + 2], a3 = anchors[4 * t + 3];
  float wa = a2 - a0, ha = a3 - a1;
  float cxa = a0 + 0.5f * wa, cya = a1 + 0.5f * ha;
  float cx = dx * wa + cxa, cy = dy * ha + cya;
  float pw = expf(dw) * wa, ph = expf(dh) * ha;
  float x1 = fminf(fmaxf(cx - 0.5f * pw, 0.f), IMG_WF);
  float x2 = fminf(fmaxf(cx + 0.5f * pw, 0.f), IMG_WF);
  float y1 = fminf(fmaxf(cy - 0.5f * ph, 0.f), IMG_HF);
  float y2 = fminf(fmaxf(cy + 0.5f * ph, 0.f), IMG_HF);
  bool ok = ((x2 - x1) >= MIN_SZ) && ((y2 - y1) >= MIN_SZ);
  cboxes[gid] = make_float4(x1, y1, x2, y2);
  ckey[gid] = ok ? sel_key[gid] : 0u;   // 0 == score -inf sentinel
  clvl[gid] = l;
}

// ---------------- K4: per-image LDS bitonic sort (descending by key) ----------------
__global__ void __launch_bounds__(1024)
k_sort(const unsigned* __restrict__ ckey, const float4* __restrict__ cboxes,
       const int* __restrict__ clvl, float4* __restrict__ sboxes,
       unsigned* __restrict__ skey, int* __restrict__ slvl) {
  int img = blockIdx.x, tid = threadIdx.x;
  __shared__ unsigned sk[SORT_N];
  __shared__ unsigned short sv[SORT_N];
  for (int i = tid; i < SORT_N; i += 1024) {
    if (i < K_ALL) { sk[i] = ckey[(size_t)img * K_ALL + i]; sv[i] = (unsigned short)i; }
    else           { sk[i] = 0u; sv[i] = 0xFFFFu; }
  }
  __syncthreads();
  for (int k = 2; k <= SORT_N; k <<= 1) {
    for (int j = k >> 1; j > 0; j >>= 1) {
      for (int i = tid; i < SORT_N; i += 1024) {
        int ixj = i ^ j;
        if (ixj > i) {
          bool dir = ((i & k) == 0);            // flipped comparison -> descending
          unsigned ki = sk[i], kj = sk[ixj];
          if ((ki < kj) == dir) {
            sk[i] = kj; sk[ixj] = ki;
            unsigned short tv = sv[i]; sv[i] = sv[ixj]; sv[ixj] = tv;
          }
        }
      }
      __syncthreads();
    }
  }
  for (int i = tid; i < K_PAD; i += 1024) {
    int v = (i < SORT_N) ? (int)sv[i] : 0xFFFF;
    float4 b = make_float4(0.f, 0.f, 0.f, 0.f);
    int l = -1; unsigned key = 0u;
    if (v < K_ALL && sk[i] != 0u) {
      b = cboxes[(size_t)img * K_ALL + v];
      l = clvl[(size_t)img * K_ALL + v];
      key = sk[i];
    }
    sboxes[(size_t)img * K_PAD + i] = b;
    skey[(size_t)img * K_PAD + i] = key;
    slvl[(size_t)img * K_PAD + i] = l;
  }
}

// ---------------- K5: row-strip IoU mask; WMMA computes area_i + area_j ----------------
// One wave per (row-tile, image); loops over column tiles cb >= rb.
// Row boxes/levels live in registers for the whole strip; A operand is loop-invariant.
__global__ void __launch_bounds__(32)
k_mask(const float4* __restrict__ sboxes, const int* __restrict__ slvl,
       unsigned short* __restrict__ mask16) {
  int rb = blockIdx.x, img = blockIdx.y;
  int lane = threadIdx.x;
  int N = lane & 15;
  int mhi = (lane >> 4) << 3;                 // rows mhi..mhi+7 for this half-wave
  size_t ibase = (size_t)img * K_PAD;

  // per-lane row data (registers, loaded once per strip)
  float4 r0, r1, r2, r3, r4, r5, r6, r7;
  int rl0, rl1, rl2, rl3, rl4, rl5, rl6, rl7;
  {
    const float4* rp = sboxes + ibase + rb * 16 + mhi;
    const int* lp = slvl + ibase + rb * 16 + mhi;
    r0 = rp[0]; r1 = rp[1]; r2 = rp[2]; r3 = rp[3];
    r4 = rp[4]; r5 = rp[5]; r6 = rp[6]; r7 = rp[7];
    rl0 = lp[0]; rl1 = lp[1]; rl2 = lp[2]; rl3 = lp[3];
    rl4 = lp[4]; rl5 = lp[5]; rl6 = lp[6]; rl7 = lp[7];
  }

  // A operand: row areas (lanes 0-15 hold rows 0..15 of this tile)
  v2f A;
  {
    float4 ab = sboxes[ibase + rb * 16 + (lane & 15)];
    float aar = (ab.z - ab.x) * (ab.w - ab.y);
    A.x = (lane < 16) ? aar : 0.f;            // K=0 (lanes 0-15), K=2 = 0
    A.y = (lane < 16) ? 1.f : 0.f;            // K=1, K=3 = 0
  }
  v8f C = {};

  for (int cb = rb; cb < NTILE; cb++) {
    // column box for this lane's N (duplicated across wave halves)
    float4 c = sboxes[ibase + cb * 16 + N];
    int cln = slvl[ibase + cb * 16 + N];
    float car = (c.z - c.x) * (c.w - c.y);

    v2f B;
    B.x = (lane < 16) ? 1.f : 0.f;
    B.y = (lane < 16) ? car : 0.f;

    v8f den;
#if __has_builtin(__builtin_amdgcn_wmma_f32_16x16x4_f32)
    den = __builtin_amdgcn_wmma_f32_16x16x4_f32(false, A, false, B, (short)0, C,
                                                false, false);
#else
    den = C; // placeholder; den stays C (all zeros)
#endif

    unsigned b0, b1, b2, b3, b4, b5, b6, b7;
#define PAIR(V, RV, RLV, BV)                                                  \
    {                                                                          \
      float iw = fmaxf(fminf(RV.z, c.z) - fmaxf(RV.x, c.x), 0.f);              \
      float ih = fmaxf(fminf(RV.w, c.w) - fmaxf(RV.y, c.y), 0.f);              \
      float inter = iw * ih;                                                   \
      float d = den[V];                                                        \
      float uni = d - inter;                                                   \
      /* inter/uni > TH  <=>  inter > TH*uni (uni>=0; uni==0 -> false = NaN) */\
      bool bit = (inter > NMS_TH * uni) && (RLV == cln) && (RLV >= 0);         \
      BV = ballot32(bit);                                                      \
    }
    PAIR(0, r0, rl0, b0) PAIR(1, r1, rl1, b1) PAIR(2, r2, rl2, b2)
    PAIR(3, r3, rl3, b3) PAIR(4, r4, rl4, b4) PAIR(5, r5, rl5, b5)
    PAIR(6, r6, rl6, b6) PAIR(7, r7, rl7, b7)
#undef PAIR

    // ballots are wave-uniform; lanes 0-15 pick their halfword with a select chain
    if (lane < 16) {
      int v = lane & 7, hh = lane >> 3;
      unsigned bv = (v == 0) ? b0 : (v == 1) ? b1 : (v == 2) ? b2 :
                    (v == 3) ? b3 : (v == 4) ? b4 : (v == 5) ? b5 :
                    (v == 6) ? b6 : b7;
      unsigned short val = hh ? (unsigned short)(bv >> 16)
                              : (unsigned short)(bv & 0xFFFFu);
      int row = rb * 16 + v + 8 * hh;
      mask16[(ibase + row) * MASK_W16 + cb] = val;
    }
  }
}

// ---------------- K6: sequential bit-scan + output (TDM loads keys to LDS) ----------------
__global__ void __launch_bounds__(256)
k_scan(const unsigned* __restrict__ skey, const float4* __restrict__ sboxes,
       const unsigned short* __restrict__ mask16, float4* __restrict__ out) {
  int img = blockIdx.x, tid = threadIdx.x;
  __shared__ unsigned skey_l[K_PAD];
  __shared__ unsigned supp[160];               // 5120 suppression bits
  __shared__ unsigned keep[1000];
  __shared__ int kcount;

#if defined(HAVE_TDM)
  if (tid == 0) {
    unsigned lds_off = (unsigned)(size_t)(&skey_l[0]);
    unsigned long long ga = (unsigned long long)(size_t)(skey + (size_t)img * K_PAD);
    tdm_load_dwords(lds_off, ga, K_PAD);
#if __has_builtin(__builtin_amdgcn_s_wait_tensorcnt)
    __builtin_amdgcn_s_wait_tensorcnt(0);
#endif
  }
#else
  for (int i = tid; i < K_PAD; i += 256) skey_l[i] = skey[(size_t)img * K_PAD + i];
#endif
  for (int i = tid; i < 160; i += 256) supp[i] = 0u;
  if (tid == 0) kcount = 0;
  __syncthreads();

  for (int i = 0; i < K_ALL; i++) {
    bool valid = (skey_l[i] != 0u);
    bool sup = ((supp[i >> 5] >> (i & 31)) & 1u) != 0u;
    if (valid && !sup) {
      if (tid == 0) {
        keep[kcount] = (unsigned)i;
        kcount++;
        __builtin_prefetch(mask16 + ((size_t)img * K_PAD + i + 1) * MASK_W16, 0, 0);
      }
      const unsigned* mrow =
          (const unsigned*)(mask16 + ((size_t)img * K_PAD + i) * MASK_W16);
      int lo = i + 1;                          // only columns j > i suppress
      for (int w = tid; w < 157; w += 256) {
        unsigned m = mrow[w];
        int base = w << 5;
        if (base + 32 <= lo) m = 0u;
        else if (base < lo) m &= ~((1u << (lo - base)) - 1u);
        supp[w] |= m;                          // unique thread per word
      }
    }
    __syncthreads();
    int kc = kcount;
    __syncthreads();
    if (kc >= 1000) break;
  }

  for (int k = tid; k < 1000; k += 256) {
    float4 b = make_float4(0.f, 0.f, 0.f, 0.f);
    if (k < kcount) b = sboxes[(size_t)img * K_PAD + keep[k]];
    out[(size_t)img * 1000 + k] = b;
  }
}

// ---------------- host launcher ----------------
extern "C" void kernel_launch(void* const* d_in, const int* in_sizes, int n_in,
                              void* d_out, int out_size, void* d_ws, size_t ws_size,
                              hipStream_t stream) {
  (void)n_in; (void)out_size;
  LevelPtrs p;
  for (int l = 0; l < 5; l++) {   // dict order interleaves obj/delta; delta is 4x bigger
    const float* a = (const float*)d_in[2 * l];
    const float* b = (const float*)d_in[2 * l + 1];
    if (in_sizes[2 * l] <= in_sizes[2 * l + 1]) { p.obj[l] = a; p.del[l] = b; }
    else                                        { p.obj[l] = b; p.del[l] = a; }
  }
  const float* anchors = (const float*)d_in[10];

  char* w = (char*)d_ws;
  size_t o = 0;
  auto carve = [&](size_t sz) -> char* {
    char* r = w + o; o += (sz + 255) & ~(size_t)255; return r;
  };
  unsigned* keys    = (unsigned*)carve((size_t)NIMG * T_ANCH * 4);
  unsigned* sel_key = (unsigned*)carve((size_t)NIMG * K_ALL * 4);
  int*      sel_idx = (int*)     carve((size_t)NIMG * K_ALL * 4);
  float4*   cboxes  = (float4*)  carve((size_t)NIMG * K_ALL * 16);
  unsigned* ckey    = (unsigned*)carve((size_t)NIMG * K_ALL * 4);
  int*      clvl    = (int*)     carve((size_t)NIMG * K_ALL * 4);
  float4*   sboxes  = (float4*)  carve((size_t)NIMG * K_PAD * 16);
  unsigned* skey    = (unsigned*)carve((size_t)NIMG * K_PAD * 4);
  int*      slvl    = (int*)     carve((size_t)NIMG * K_PAD * 4);
  unsigned short* mask16 = (unsigned short*)carve((size_t)NIMG * K_PAD * MASK_W16 * 2);
  if (o > ws_size) return;   // not enough scratch: do nothing deterministically

  int total = NIMG * T_ANCH;
  k_scores<<<(total + 255) / 256, 256, 0, stream>>>(p, keys);
  k_select<<<dim3(NIMG, 5), 256, 0, stream>>>(keys, sel_key, sel_idx);
  k_gather<<<(NIMG * K_ALL + 255) / 256, 256, 0, stream>>>(p, anchors, sel_key, sel_idx,
                                                           cboxes, ckey, clvl);
  k_sort<<<NIMG, 1024, 0, stream>>>(ckey, cboxes, clvl, sboxes, skey, slvl);
  k_mask<<<dim3(NTILE, NIMG), 32, 0, stream>>>(sboxes, slvl, mask16);
  k_scan<<<NIMG, 256, 0, stream>>>(skey, sboxes, mask16, (float4*)d_out);
}